// STULayer_48550310314365
// MI455X (gfx1250) — compile-verified
//
#include <hip/hip_runtime.h>
#include <hip/hip_bf16.h>

// ---- problem constants -----------------------------------------------------
constexpr int kB  = 8;
constexpr int kD  = 128;
constexpr int kG  = 2048;
constexpr int kL  = 24;
constexpr int kGF = kG / 2 + 1;          // 1025 rfft bins
constexpr int kNF = 2176;                // padded 2*kGF (re|im concat), multiple of 128
constexpr int kNROW = kB * kD;           // 1024 rows for the batched DFT GEMMs

typedef float v2f __attribute__((ext_vector_type(2)));
typedef float v4f __attribute__((ext_vector_type(4)));
typedef float v8f __attribute__((ext_vector_type(8)));

// fp32 WMMA: D(16x16) = A(16x4) * B(4x16) + C, wave32.
__device__ __forceinline__ v8f wmma4(v2f a, v2f b, v8f c) {
  return __builtin_amdgcn_wmma_f32_16x16x4_f32(false, a, false, b, (short)0, c,
                                               false, false);
}

__device__ __forceinline__ v8f zero8() {
  v8f z = {0.f, 0.f, 0.f, 0.f, 0.f, 0.f, 0.f, 0.f};
  return z;
}

// ---- CDNA5 async copy: global -> LDS, 16B per lane, tracked by ASYNCcnt ----
// (CDNA5 ISA / CDNA5 ISA: GLOBAL_LOAD_ASYNC_TO_LDS_B128,
//  GV mode: 64-bit VGPR address, SADDR=off. LDS dest addr = generic VA[31:0].)
__device__ __forceinline__ void async_copy16(const float* g, float* l) {
  unsigned loff = (unsigned)(unsigned long long)(uintptr_t)l;   // LDS byte addr
  unsigned long long ga = (unsigned long long)(uintptr_t)g;
  asm volatile("global_load_async_to_lds_b128 %0, %1, off"
               :: "v"(loff), "v"(ga) : "memory");
}
__device__ __forceinline__ void wait_async0() {
  asm volatile("s_wait_asynccnt 0x0" ::: "memory");
}
__device__ __forceinline__ void wait_async8() {
  asm volatile("s_wait_asynccnt 0x8" ::: "memory");
}

// ---- kernel 0: build real-DFT matrices ------------------------------------
__global__ void __launch_bounds__(256) stu_gen_dft(float* __restrict__ dfwd,
                                                   float* __restrict__ dinv) {
  int idx = blockIdx.x * blockDim.x + threadIdx.x;
  if (idx >= kG * kNF) return;
  int g = idx / kNF, fcol = idx % kNF;
  float vf = 0.f, vi = 0.f;
  if (fcol < 2 * kGF) {
    int f = (fcol < kGF) ? fcol : fcol - kGF;
    int r = (g * f) & (kG - 1);                       // exact angle reduction
    float ang = (6.283185307179586f / (float)kG) * (float)r;
    float s, c;
    __sincosf(ang, &s, &c);
    float w = (f == 0 || f == kG / 2) ? (1.0f / kG) : (2.0f / kG);
    if (fcol < kGF) { vf = c;  vi = w * c; }
    else            { vf = -s; vi = -w * s; }
  }
  dfwd[(size_t)g * kNF + fcol] = vf;
  dinv[(size_t)fcol * kG + g]  = vi;
}

// ---- kernel 1: LayerNorm over channel dim D -------------------------------
__global__ void __launch_bounds__(256) stu_layernorm(
    const float* __restrict__ x, const float* __restrict__ gamma,
    const float* __restrict__ beta, float* __restrict__ z) {
  int g = blockIdx.x * blockDim.x + threadIdx.x;
  int b = blockIdx.y;
  const float* xp = x + ((size_t)b * kD) * kG + g;
  float sum = 0.f, sq = 0.f;
  for (int d = 0; d < kD; ++d) { float v = xp[(size_t)d * kG]; sum += v; sq += v * v; }
  float mean = sum * (1.0f / kD);
  float var  = sq * (1.0f / kD) - mean * mean;
  float rstd = rsqrtf(var + 1e-5f);
  float* zp = z + ((size_t)b * kD) * kG + g;
  for (int d = 0; d < kD; ++d) {
    float v = xp[(size_t)d * kG];
    zp[(size_t)d * kG] = (v - mean) * rstd * gamma[d] + beta[d];
  }
}

// ---- fp32 WMMA GEMM with async double-buffered LDS staging ----------------
// C(MxN) = A(MxK) @ B(KxN), row-major. Block tile 128x128, K-step 32,
// 256 threads = 8 wave32s; wave owns 32x64 (2x4 accum tiles of 16x16).
// Dynamic LDS: As[2] stride 36, Bs[2] stride 128  -> 69632 bytes.
__global__ void __launch_bounds__(256) stu_gemm_f32(
    const float* __restrict__ A, const float* __restrict__ Bm,
    float* __restrict__ C, int K, int lda, int ldb, int ldc) {
  extern __shared__ float smem[];
  float* As0 = smem;                    // 128*36
  float* As1 = As0 + 128 * 36;
  float* Bs0 = As1 + 128 * 36;          // 32*128
  float* Bs1 = Bs0 + 32 * 128;

  const int tid  = threadIdx.x;
  const int wave = tid >> 5, lane = tid & 31;
  const int lr   = lane & 15, lh = lane >> 4;
  const int m0 = blockIdx.y * 128, n0 = blockIdx.x * 128;
  const int r0 = (wave & 3) * 32, c0 = (wave >> 2) * 64;

  v8f acc[2][4];
  for (int rt = 0; rt < 2; ++rt)
    for (int ct = 0; ct < 4; ++ct) acc[rt][ct] = zero8();

  // issue one tile of async copies: 4x A (16B) + 4x B (16B) per thread
  auto stage = [&](int kb, int buf) {
    float* Asb = buf ? As1 : As0;
    float* Bsb = buf ? Bs1 : Bs0;
    const float* Ag = A + (size_t)m0 * lda + kb;
#pragma unroll
    for (int i = 0; i < 4; ++i) {
      int e = tid + i * 256;                 // 1024 x 16B  (128x32 tile)
      int row = e >> 3, c4 = (e & 7) << 2;
      async_copy16(Ag + (size_t)row * lda + c4, Asb + row * 36 + c4);
    }
    const float* Bg = Bm + (size_t)kb * ldb + n0;
#pragma unroll
    for (int i = 0; i < 4; ++i) {
      int e = tid + i * 256;                 // 1024 x 16B  (32x128 tile)
      int row = e >> 5, c4 = (e & 31) << 2;
      async_copy16(Bg + (size_t)row * ldb + c4, Bsb + row * 128 + c4);
    }
  };

  const int nk = K >> 5;
  stage(0, 0);
  for (int t = 0; t < nk; ++t) {
    const int buf = t & 1;
    if (t + 1 < nk) {
      stage((t + 1) << 5, buf ^ 1);   // prefetch next tile into other buffer
      wait_async8();                  // current tile's 8 ops retired (in-order)
    } else {
      wait_async0();
    }
    __syncthreads();

    const float* Asb = buf ? As1 : As0;
    const float* Bsb = buf ? Bs1 : Bs0;
#pragma unroll
    for (int ks = 0; ks < 8; ++ks) {
      int kk = ks * 4 + 2 * lh;
      v2f a0 = *(const v2f*)(Asb + (r0 + lr) * 36 + kk);
      v2f a1 = *(const v2f*)(Asb + (r0 + 16 + lr) * 36 + kk);
#pragma unroll
      for (int ct = 0; ct < 4; ++ct) {
        int cc = c0 + ct * 16 + lr;
        v2f bf;
        bf.x = Bsb[kk * 128 + cc];
        bf.y = Bsb[(kk + 1) * 128 + cc];
        acc[0][ct] = wmma4(a0, bf, acc[0][ct]);
        acc[1][ct] = wmma4(a1, bf, acc[1][ct]);
      }
    }
    __syncthreads();   // all waves done reading buf before it is re-staged
  }

  for (int rt = 0; rt < 2; ++rt)
    for (int ct = 0; ct < 4; ++ct)
#pragma unroll
      for (int i = 0; i < 8; ++i) {
        int row = m0 + r0 + rt * 16 + i + 8 * lh;
        int col = n0 + c0 + ct * 16 + lr;
        C[(size_t)row * ldc + col] = acc[rt][ct][i];
      }
}

// ---- kernel 3: spectral mix -----------------------------------------------
// Sf[b,h,fcol] = sum_l sum_d Theta[l,h,d] * Xf[b,d,fcol] * phi(l,fcol)
// phi folded into B fragments so WMMA C-chains across k AND l.
__global__ void __launch_bounds__(256) stu_spectral_mix(
    const float* __restrict__ Xf, const float* __restrict__ Theta,
    const float* __restrict__ Phi, float* __restrict__ Sf) {
  __shared__ float Xs[128 * 68];               // 128 d x 64 freq-cols
  const int tid  = threadIdx.x;
  const int wave = tid >> 5, lane = tid & 31;
  const int lr   = lane & 15, lh = lane >> 4;
  const int b = blockIdx.y;
  const int col0 = blockIdx.x * 64;

#pragma unroll
  for (int i = 0; i < 8; ++i) {                // 2048 x 16B async into LDS
    int e = tid + i * 256;
    int row = e >> 4, c4 = (e & 15) << 2;
    async_copy16(Xf + (size_t)(b * kD + row) * kNF + col0 + c4,
                 Xs + row * 68 + c4);
  }
  wait_async0();
  __syncthreads();

  const int h0 = wave * 16;                    // 8 waves x 16 rows = 128 h
  v8f acc[4];
  for (int ct = 0; ct < 4; ++ct) acc[ct] = zero8();

  for (int l = 0; l < kL; ++l) {
    float phi[4];
#pragma unroll
    for (int ct = 0; ct < 4; ++ct) {
      int col = col0 + ct * 16 + lr;
      float p = 0.f;
      if (col < kGF)          p = Phi[(size_t)l * kGF + col];
      else if (col < 2 * kGF) p = Phi[(size_t)l * kGF + (col - kGF)];
      phi[ct] = p;
    }
    const float* Tl = Theta + (size_t)l * kD * kD + (size_t)(h0 + lr) * kD;
#pragma unroll 4
    for (int ks = 0; ks < 32; ++ks) {
      int kk = ks * 4 + 2 * lh;
      v2f a = *(const v2f*)(Tl + kk);
#pragma unroll
      for (int ct = 0; ct < 4; ++ct) {
        int cc = ct * 16 + lr;
        v2f bf;
        bf.x = Xs[kk * 68 + cc] * phi[ct];
        bf.y = Xs[(kk + 1) * 68 + cc] * phi[ct];
        acc[ct] = wmma4(a, bf, acc[ct]);
      }
    }
  }
  for (int ct = 0; ct < 4; ++ct)
#pragma unroll
    for (int i = 0; i < 8; ++i) {
      int row = b * kD + h0 + i + 8 * lh;
      int col = col0 + ct * 16 + lr;
      Sf[(size_t)row * kNF + col] = acc[ct][i];
    }
}

// ---- kernel 5: fused pointwise MLP + residual -----------------------------
__global__ void __launch_bounds__(256) stu_mlp(
    const float* __restrict__ S, const float* __restrict__ x,
    const float* __restrict__ fc1w, const float* __restrict__ fc1b,
    const float* __restrict__ fc2w, const float* __restrict__ fc2b,
    float* __restrict__ out) {
  __shared__ float Ss[128 * 36];               // 128 h  x 32 g
  __shared__ float Hs[256 * 36];               // 256 h' x 32 g
  const int tid  = threadIdx.x;
  const int wave = tid >> 5, lane = tid & 31;
  const int lr   = lane & 15, lh = lane >> 4;
  const int b = blockIdx.y;
  const int g0 = blockIdx.x * 32;

#pragma unroll
  for (int i = 0; i < 4; ++i) {                // 1024 x 16B async into LDS
    int e = tid + i * 256;
    int row = e >> 3, c4 = (e & 7) << 2;
    async_copy16(S + (size_t)(b * kD + row) * kG + g0 + c4,
                 Ss + row * 36 + c4);
  }
  wait_async0();
  __syncthreads();

  // phase 1: H(256x32) = fc1w(256x128) @ Ss ; wave owns rows [wave*32, +32)
  {
    int r0 = wave * 32;
    v8f acc[2][2];
    for (int rt = 0; rt < 2; ++rt)
      for (int ct = 0; ct < 2; ++ct) acc[rt][ct] = zero8();
#pragma unroll 4
    for (int ks = 0; ks < 32; ++ks) {
      int kk = ks * 4 + 2 * lh;
      v2f a0 = *(const v2f*)(fc1w + (size_t)(r0 + lr) * kD + kk);
      v2f a1 = *(const v2f*)(fc1w + (size_t)(r0 + 16 + lr) * kD + kk);
#pragma unroll
      for (int ct = 0; ct < 2; ++ct) {
        int cc = ct * 16 + lr;
        v2f bf;
        bf.x = Ss[kk * 36 + cc];
        bf.y = Ss[(kk + 1) * 36 + cc];
        acc[0][ct] = wmma4(a0, bf, acc[0][ct]);
        acc[1][ct] = wmma4(a1, bf, acc[1][ct]);
      }
    }
    for (int rt = 0; rt < 2; ++rt)
      for (int ct = 0; ct < 2; ++ct)
#pragma unroll
        for (int i = 0; i < 8; ++i) {
          int row = r0 + rt * 16 + i + 8 * lh;
          int col = ct * 16 + lr;
          float h = acc[rt][ct][i] + fc1b[row];
          h = 0.5f * h * (1.0f + erff(h * 0.7071067811865475f));   // exact gelu
          Hs[row * 36 + col] = h;
        }
  }
  __syncthreads();

  // phase 2: out(128x32) = fc2w(128x256) @ Hs ; wave owns row-tile wave*16
  {
    int d0 = wave * 16;
    v8f acc[2];
    acc[0] = zero8(); acc[1] = zero8();
#pragma unroll 4
    for (int ks = 0; ks < 64; ++ks) {
      int kk = ks * 4 + 2 * lh;
      v2f a = *(const v2f*)(fc2w + (size_t)(d0 + lr) * (2 * kD) + kk);
#pragma unroll
      for (int ct = 0; ct < 2; ++ct) {
        int cc = ct * 16 + lr;
        v2f bf;
        bf.x = Hs[kk * 36 + cc];
        bf.y = Hs[(kk + 1) * 36 + cc];
        acc[ct] = wmma4(a, bf, acc[ct]);
      }
    }
    for (int ct = 0; ct < 2; ++ct)
#pragma unroll
      for (int i = 0; i < 8; ++i) {
        int row = d0 + i + 8 * lh;
        int col = g0 + ct * 16 + lr;
        size_t idx = ((size_t)b * kD + row) * kG + col;
        out[idx] = x[idx] + acc[ct][i] + fc2b[row];
      }
  }
}

// ---- host-side launch ------------------------------------------------------
extern "C" void kernel_launch(void* const* d_in, const int* in_sizes, int n_in,
                              void* d_out, int out_size, void* d_ws, size_t ws_size,
                              hipStream_t stream) {
  (void)in_sizes; (void)n_in; (void)out_size; (void)ws_size;
  const float* x     = (const float*)d_in[0];
  const float* Phi   = (const float*)d_in[1];
  const float* Theta = (const float*)d_in[2];
  const float* gam   = (const float*)d_in[3];
  const float* bet   = (const float*)d_in[4];
  const float* fc1w  = (const float*)d_in[5];
  const float* fc1b  = (const float*)d_in[6];
  const float* fc2w  = (const float*)d_in[7];
  const float* fc2b  = (const float*)d_in[8];
  float* out = (float*)d_out;

  float* ws   = (float*)d_ws;                         // ~70 MB total
  float* z    = ws;                                   // 1024 x 2048
  float* Xf   = z    + (size_t)kNROW * kG;            // 1024 x 2176
  float* Sf   = Xf   + (size_t)kNROW * kNF;           // 1024 x 2176
  float* S    = Sf   + (size_t)kNROW * kNF;           // 1024 x 2048
  float* Dfwd = S    + (size_t)kNROW * kG;            // 2048 x 2176
  float* Dinv = Dfwd + (size_t)kG * kNF;              // 2176 x 2048

  const size_t gemm_lds = (2 * 128 * 36 + 2 * 32 * 128) * sizeof(float); // 69632

  stu_gen_dft<<<(kG * kNF + 255) / 256, 256, 0, stream>>>(Dfwd, Dinv);
  stu_layernorm<<<dim3(kG / 256, kB), 256, 0, stream>>>(x, gam, bet, z);
  // Xf = z @ Dfwd   (M=1024, N=2176, K=2048)
  stu_gemm_f32<<<dim3(kNF / 128, kNROW / 128), 256, gemm_lds, stream>>>(
      z, Dfwd, Xf, kG, kG, kNF, kNF);
  // Sf = sum_l Theta_l @ (Xf * phi_l)
  stu_spectral_mix<<<dim3(kNF / 64, kB), 256, 0, stream>>>(Xf, Theta, Phi, Sf);
  // S = Sf @ Dinv   (M=1024, N=2048, K=2176)
  stu_gemm_f32<<<dim3(kG / 128, kNROW / 128), 256, gemm_lds, stream>>>(
      Sf, Dinv, S, kNF, kNF, kG, kG);
  // out = x + MLP(S)
  stu_mlp<<<dim3(kG / 32, kB), 256, 0, stream>>>(S, x, fc1w, fc1b, fc2w, fc2b, out);
}